// AlphaFoldModel_80393197846632
// MI455X (gfx1250) — compile-verified
//
#include <hip/hip_runtime.h>
#include <math.h>

// ---------------------------------------------------------------------------
// Mini-AlphaFold forward for MI455X (gfx1250, wave32, WMMA).
// All GEMMs go through a fp32 WMMA 16x16x4 kernel (exact fp32 like the
// reference). One wave computes a 32x32 tile with 4 accumulators: per K-step
// of 4 -> 4x v2f (b64) loads feeding 4 v_wmma, with A/B fragment reuse.
// Edge tiles use a branchless clamped-address + select path (no exec-mask
// branches). Softmax/attention/IPA glue are scalar VALU kernels.
// ---------------------------------------------------------------------------

typedef __attribute__((ext_vector_type(2))) float v2f;
typedef __attribute__((ext_vector_type(8))) float v8f;

static inline int cdiv(int a, int b) { return (a + b - 1) / b; }

__device__ __forceinline__ v8f wmma4(v2f a, v2f b, v8f c) {
  return __builtin_amdgcn_wmma_f32_16x16x4_f32(
      /*neg_a=*/false, a, /*neg_b=*/false, b,
      /*c_mod=*/(short)0, c, /*reuse_a=*/false, /*reuse_b=*/false);
}

// ------------------------- WMMA fp32 GEMM (A @ W^T) ------------------------
// C[M,N] = act(A[M,K] @ W[N,K]^T + bias[N]) (+ res[M,N] if RES)
// One wave per 32x32 tile. A frag (16x4 f32): lanes 0-15 -> M=lane, K=k0+{0,1};
// lanes 16-31 -> M=lane-16, K=k0+{2,3}. C/D: VGPR g -> row g (lanes 0-15) /
// 8+g (lanes 16-31), col = lane&15.
template<int ACT, int RES>
__global__ __launch_bounds__(32)
void gemm_nt_wmma(const float* __restrict__ A, const float* __restrict__ W,
                  const float* __restrict__ bias, const float* __restrict__ res,
                  float* __restrict__ C, int M, int N, int K)
{
  const int tn   = blockIdx.x * 32;
  const int tm   = blockIdx.y * 32;
  const int lane = threadIdx.x;
  const int half = lane >> 4;
  const int l16  = lane & 15;

  const int am0 = tm + l16;       // rows for acc0x
  const int am1 = am0 + 16;       // rows for acc1x
  const int bn0 = tn + l16;       // cols for accx0
  const int bn1 = bn0 + 16;       // cols for accx1

  v8f acc00 = {}, acc01 = {}, acc10 = {}, acc11 = {};

  const bool full = (tm + 32 <= M) && (tn + 32 <= N);
  int k0 = 0;

  if (full && ((K & 1) == 0)) {
    // -------- fast path: unconditional 8-byte vector loads --------
    const v2f* A0 = (const v2f*)(A + (size_t)am0 * K);
    const v2f* A1 = (const v2f*)(A + (size_t)am1 * K);
    const v2f* W0 = (const v2f*)(W + (size_t)bn0 * K);
    const v2f* W1 = (const v2f*)(W + (size_t)bn1 * K);
#pragma unroll 2
    for (; k0 + 4 <= K; k0 += 4) {
      const int kp = (k0 >> 1) + half;   // v2f index: K pair for this half-wave
      const v2f a0 = A0[kp];
      const v2f a1 = A1[kp];
      const v2f b0 = W0[kp];
      const v2f b1 = W1[kp];
      acc00 = wmma4(a0, b0, acc00);
      acc01 = wmma4(a0, b1, acc01);
      acc10 = wmma4(a1, b0, acc10);
      acc11 = wmma4(a1, b1, acc11);
    }
  }

  // -------- guarded path (edge tiles / K remainder): branchless -----------
  // Clamp addresses (always legal), select values to zero. No exec branches.
  if (k0 < K) {
    const float* Ar0 = A + (size_t)(am0 < M ? am0 : 0) * K;
    const float* Ar1 = A + (size_t)(am1 < M ? am1 : 0) * K;
    const float* Wr0 = W + (size_t)(bn0 < N ? bn0 : 0) * K;
    const float* Wr1 = W + (size_t)(bn1 < N ? bn1 : 0) * K;
    const bool m0v = (am0 < M), m1v = (am1 < M);
    const bool n0v = (bn0 < N), n1v = (bn1 < N);
    for (; k0 < K; k0 += 4) {
      const int ka  = k0 + half * 2;
      const int kaA = (ka     < K) ? ka     : 0;
      const int kaB = (ka + 1 < K) ? ka + 1 : 0;
      const bool kva = (ka < K), kvb = (ka + 1 < K);
      v2f a0, a1, b0, b1;
      { float x = Ar0[kaA], y = Ar0[kaB];
        a0.x = (m0v && kva) ? x : 0.f;  a0.y = (m0v && kvb) ? y : 0.f; }
      { float x = Ar1[kaA], y = Ar1[kaB];
        a1.x = (m1v && kva) ? x : 0.f;  a1.y = (m1v && kvb) ? y : 0.f; }
      { float x = Wr0[kaA], y = Wr0[kaB];
        b0.x = (n0v && kva) ? x : 0.f;  b0.y = (n0v && kvb) ? y : 0.f; }
      { float x = Wr1[kaA], y = Wr1[kaB];
        b1.x = (n1v && kva) ? x : 0.f;  b1.y = (n1v && kvb) ? y : 0.f; }
      acc00 = wmma4(a0, b0, acc00);
      acc01 = wmma4(a0, b1, acc01);
      acc10 = wmma4(a1, b0, acc10);
      acc11 = wmma4(a1, b1, acc11);
    }
  }

  // ------------------------------- store ----------------------------------
  auto emit = [&](v8f acc, int bm, int cn) {
    if (cn < N) {
      const float bv = bias ? bias[cn] : 0.f;
#pragma unroll
      for (int g = 0; g < 8; ++g) {
        const int cm = bm + g + 8 * half;
        if (cm < M) {
          float r = acc[g] + bv;
          if (ACT == 1) r = fmaxf(r, 0.f);
          if (ACT == 2) r = 1.f / (1.f + __expf(-r));
          const size_t o = (size_t)cm * N + cn;
          if (RES) r += res[o];
          C[o] = r;
        }
      }
    }
  };
  emit(acc00, tm,      bn0);
  emit(acc01, tm,      bn1);
  emit(acc10, tm + 16, bn0);
  emit(acc11, tm + 16, bn1);
}

static void gemm(hipStream_t st, const float* A, const float* W, const float* bias,
                 const float* res, float* C, int M, int N, int K, int act)
{
  dim3 g(cdiv(N, 32), cdiv(M, 32)), b(32);
  if (res) {
    if (act == 0)      gemm_nt_wmma<0,1><<<g,b,0,st>>>(A,W,bias,res,C,M,N,K);
    else if (act == 1) gemm_nt_wmma<1,1><<<g,b,0,st>>>(A,W,bias,res,C,M,N,K);
    else               gemm_nt_wmma<2,1><<<g,b,0,st>>>(A,W,bias,res,C,M,N,K);
  } else {
    if (act == 0)      gemm_nt_wmma<0,0><<<g,b,0,st>>>(A,W,bias,res,C,M,N,K);
    else if (act == 1) gemm_nt_wmma<1,0><<<g,b,0,st>>>(A,W,bias,res,C,M,N,K);
    else               gemm_nt_wmma<2,0><<<g,b,0,st>>>(A,W,bias,res,C,M,N,K);
  }
}

// ----------------------- MHA (online-softmax, scalar) ----------------------
// qkv rows: [(n*L+i), 3*E]  (q | k | v).  One thread per (n, h, i).
template<int DH>
__global__ void mha_attn(const float* __restrict__ qkv, float* __restrict__ out,
                         int Nb, int L, int H, float scale)
{
  const int idx = blockIdx.x * blockDim.x + threadIdx.x;
  if (idx >= Nb * H * L) return;
  const int i = idx % L;
  const int h = (idx / L) % H;
  const int n = idx / (L * H);
  const int E = H * DH;
  const size_t row3 = (size_t)3 * E;

  const float* q = qkv + (size_t)(n * L + i) * row3 + h * DH;
  float qr[DH], acc[DH];
#pragma unroll
  for (int d = 0; d < DH; ++d) { qr[d] = q[d]; acc[d] = 0.f; }

  float m = -3.0e38f, l = 0.f;
  for (int j = 0; j < L; ++j) {
    const float* kr = qkv + (size_t)(n * L + j) * row3 + E + h * DH;
    float s = 0.f;
#pragma unroll
    for (int d = 0; d < DH; ++d) s += qr[d] * kr[d];
    s *= scale;
    const float mn = fmaxf(m, s);
    const float c  = __expf(m - mn);
    const float p  = __expf(s - mn);
    l = l * c + p;
    const float* vr = kr + E;
#pragma unroll
    for (int d = 0; d < DH; ++d) acc[d] = acc[d] * c + p * vr[d];
    m = mn;
  }
  const float inv = 1.f / l;
  float* o = out + (size_t)(n * L + i) * E + h * DH;
#pragma unroll
  for (int d = 0; d < DH; ++d) o[d] = acc[d] * inv;
}

// ------------------------------- LayerNorm ---------------------------------
__global__ void layernorm_rows(const float* __restrict__ x, const float* __restrict__ g,
                               const float* __restrict__ b, float* __restrict__ y,
                               int rows, int D)
{
  const int r = blockIdx.x * blockDim.x + threadIdx.x;
  if (r >= rows) return;
  const float* xr = x + (size_t)r * D;
  float* yr = y + (size_t)r * D;
  float mu = 0.f;
  for (int d = 0; d < D; ++d) mu += xr[d];
  mu /= (float)D;
  float var = 0.f;
  for (int d = 0; d < D; ++d) { const float t = xr[d] - mu; var += t * t; }
  var /= (float)D;
  const float inv = rsqrtf(var + 1e-5f);
  for (int d = 0; d < D; ++d) yr[d] = (xr[d] - mu) * inv * g[d] + b[d];
}

// --------------------------- small elementwise -----------------------------
__global__ void tr_ns(const float* __restrict__ x, float* __restrict__ y,
                      int N, int S_, int D)  // y[s,n,d] = x[n,s,d]
{
  const int idx = blockIdx.x * blockDim.x + threadIdx.x;
  if (idx >= N * S_ * D) return;
  const int d = idx % D;
  const int r = idx / D;
  const int n = r % N;
  const int s = r / N;
  y[idx] = x[((size_t)n * S_ + s) * D + d];
}

__global__ void add_tr(float* __restrict__ msa, const float* __restrict__ t,
                       int N, int S_, int D)  // msa[n,s,d] += t[s,n,d]
{
  const int idx = blockIdx.x * blockDim.x + threadIdx.x;
  if (idx >= N * S_ * D) return;
  const int d = idx % D;
  const int r = idx / D;
  const int s = r % S_;
  const int n = r / S_;
  msa[idx] += t[((size_t)s * N + n) * D + d];
}

__global__ void mean_seq(const float* __restrict__ msa, float* __restrict__ mp,
                         int N, int S_, int D)
{
  const int idx = blockIdx.x * blockDim.x + threadIdx.x;
  if (idx >= S_ * D) return;
  const int d = idx % D;
  const int s = idx / D;
  float sum = 0.f;
  for (int n = 0; n < N; ++n) sum += msa[((size_t)n * S_ + s) * D + d];
  mp[idx] = sum / (float)N;
}

__global__ void outer_k(const float* __restrict__ mp, float* __restrict__ o,
                        int S_, int D)  // o[(i*S+j), k] = mp[i,k]*mp[j,k]
{
  const int idx = blockIdx.x * blockDim.x + threadIdx.x;
  if (idx >= S_ * S_ * D) return;
  const int k = idx % D;
  const int r = idx / D;
  const int j = r % S_;
  const int i = r / S_;
  o[idx] = mp[(size_t)i * D + k] * mp[(size_t)j * D + k];
}

__global__ void copy_k(const float* __restrict__ a, float* __restrict__ b, int n)
{
  const int idx = blockIdx.x * blockDim.x + threadIdx.x;
  if (idx < n) b[idx] = a[idx];
}

// ------------------------------- IPA pieces --------------------------------
__global__ void frame_init_k(float* __restrict__ F, int S_)
{
  const int idx = blockIdx.x * blockDim.x + threadIdx.x;
  if (idx >= S_ * 12) return;
  const int c = idx % 12;
  F[idx] = (c == 0 || c == 4 || c == 8) ? 1.f : 0.f;
}

// G[s, hp, c] = sum_k P[s,hp,k]*R[s,k,c] + t[s,c]   (frames: [t(3) | R(9)])
__global__ void frame_apply_k(const float* __restrict__ P, const float* __restrict__ F,
                              float* __restrict__ G, int S_, int HP)
{
  const int idx = blockIdx.x * blockDim.x + threadIdx.x;
  if (idx >= S_ * HP * 3) return;
  const int c  = idx % 3;
  const int hp = (idx / 3) % HP;
  const int s  = idx / (3 * HP);
  const float* f  = F + (size_t)s * 12;
  const float* pr = P + ((size_t)s * HP + hp) * 3;
  float sum = f[c];
  for (int k = 0; k < 3; ++k) sum += pr[k] * f[3 + k * 3 + c];
  G[idx] = sum;
}

__global__ void ipa_logits_k(const float* __restrict__ q, const float* __restrict__ k,
                             const float* __restrict__ qg, const float* __restrict__ kg,
                             const float* __restrict__ ebb, const float* __restrict__ gamma,
                             float* __restrict__ lg, int S_, int H, int D, int HP3, float scale)
{
  const int idx = blockIdx.x * blockDim.x + threadIdx.x;
  if (idx >= S_ * S_ * H) return;
  const int h = idx % H;
  const int j = (idx / H) % S_;
  const int i = idx / (H * S_);
  const float* qi = q + ((size_t)i * H + h) * D;
  const float* kj = k + ((size_t)j * H + h) * D;
  float s = 0.f;
  for (int d = 0; d < D; ++d) s += qi[d] * kj[d];
  s *= scale;
  const float* qgi = qg + (size_t)i * HP3 + h * 12;
  const float* kgj = kg + (size_t)j * HP3 + h * 12;
  float pt = 0.f;
  for (int c = 0; c < 12; ++c) { const float df = qgi[c] - kgj[c]; pt += df * df; }
  lg[idx] = s - gamma[0] * pt + ebb[((size_t)i * S_ + j) * H + h];
}

__global__ void softmax_over_j(const float* __restrict__ lg, float* __restrict__ a,
                               int S_, int H)
{
  const int idx = blockIdx.x * blockDim.x + threadIdx.x;
  if (idx >= S_ * H) return;
  const int h = idx % H;
  const int i = idx / H;
  const float* base = lg + (size_t)i * S_ * H + h;
  float* ab = a + (size_t)i * S_ * H + h;
  float m = -3.0e38f;
  for (int j = 0; j < S_; ++j) m = fmaxf(m, base[(size_t)j * H]);
  float sum = 0.f;
  for (int j = 0; j < S_; ++j) sum += __expf(base[(size_t)j * H] - m);
  const float inv = 1.f / sum;
  for (int j = 0; j < S_; ++j) ab[(size_t)j * H] = __expf(base[(size_t)j * H] - m) * inv;
}

__global__ void ipa_so_k(const float* __restrict__ a, const float* __restrict__ v,
                         float* __restrict__ cat, int S_, int H, int D, int CATN)
{
  const int idx = blockIdx.x * blockDim.x + threadIdx.x;
  if (idx >= S_ * H * D) return;
  const int d = idx % D;
  const int h = (idx / D) % H;
  const int i = idx / (D * H);
  float s = 0.f;
  for (int j = 0; j < S_; ++j)
    s += a[((size_t)i * S_ + j) * H + h] * v[((size_t)j * H + h) * D + d];
  cat[(size_t)i * CATN + h * D + d] = s;
}

__global__ void ipa_po_k(const float* __restrict__ a, const float* __restrict__ vp,
                         float* __restrict__ cat, int S_, int H, int HP3, int CATN, int off)
{
  const int idx = blockIdx.x * blockDim.x + threadIdx.x;
  if (idx >= S_ * HP3) return;
  const int c = idx % HP3;
  const int i = idx / HP3;
  const int h = c / 12;
  float s = 0.f;
  for (int j = 0; j < S_; ++j)
    s += a[((size_t)i * S_ + j) * H + h] * vp[(size_t)j * HP3 + c];
  cat[(size_t)i * CATN + off + c] = s;
}

__global__ void frame_update_k(float* __restrict__ F, const float* __restrict__ fu, int S_)
{
  const int s = blockIdx.x * blockDim.x + threadIdx.x;
  if (s >= S_) return;
  float* f = F + (size_t)s * 12;
  const float* u = fu + (size_t)s * 12;
  float nf[12];
  nf[0] = f[0] + u[0]; nf[1] = f[1] + u[1]; nf[2] = f[2] + u[2];
  for (int i = 0; i < 3; ++i)
    for (int k = 0; k < 3; ++k) {
      float sum = 0.f;
      for (int j = 0; j < 3; ++j) sum += f[3 + i * 3 + j] * u[3 + j * 3 + k];
      nf[3 + i * 3 + k] = sum;
    }
  for (int c = 0; c < 12; ++c) f[c] = nf[c];
}

// ------------------------- parameter leaf parsing --------------------------
// JAX pytree order: dict keys sorted alphabetically, lists in order.
struct LinP { const float* b; const float* w; };
struct MhaP { const float* in_b; const float* in_w; LinP out; };
struct TrP  { LinP l1; LinP l2; const float* ln_b; const float* ln_g; };
struct BlkP { MhaP col; LinP m2p; TrP msa_tr; MhaP pair_attn; TrP pair_tr; MhaP row; };
struct IpaP { LinP eb; const float* gamma; LinP k; LinP kp; LinP out; LinP q; LinP qp; LinP v; LinP vp; };
struct StrP { LinP fu1; LinP fu2; IpaP ipa; };

static LinP parse_lin(void* const* d, int& i) {
  LinP L; L.b = (const float*)d[i++]; L.w = (const float*)d[i++]; return L;
}
static MhaP parse_mha(void* const* d, int& i) {
  MhaP M; M.in_b = (const float*)d[i++]; M.in_w = (const float*)d[i++];
  M.out = parse_lin(d, i); return M;
}
static TrP parse_tr(void* const* d, int& i) {
  TrP T; T.l1 = parse_lin(d, i); T.l2 = parse_lin(d, i);
  T.ln_b = (const float*)d[i++]; T.ln_g = (const float*)d[i++]; return T;
}

// --------------------------------- driver ----------------------------------
extern "C" void kernel_launch(void* const* d_in, const int* in_sizes, int n_in,
                              void* d_out, int out_size, void* d_ws, size_t ws_size,
                              hipStream_t stream)
{
  (void)in_sizes; (void)n_in; (void)out_size; (void)ws_size;
  constexpr int NSEQ = 16, S = 48, MSA_D = 256, PAIR_D = 128, NODE_D = 384;
  constexpr int EVO_H = 8, IPA_H = 12, IPA_P = 4, VOCAB = 21;
  constexpr int MR = NSEQ * S;          // 768 MSA rows
  constexpr int PR = S * S;             // 2304 pair rows
  constexpr int HP3 = IPA_H * IPA_P * 3;        // 144
  constexpr int CATN = IPA_H * NODE_D + HP3;    // 4752

  int ii = 0;
  const float* msa_features  = (const float*)d_in[ii++];
  const float* pair_features = (const float*)d_in[ii++];
  BlkP blk[2];
  for (int b = 0; b < 2; ++b) {
    blk[b].col = parse_mha(d_in, ii);
    blk[b].m2p = parse_lin(d_in, ii);
    blk[b].msa_tr = parse_tr(d_in, ii);
    blk[b].pair_attn = parse_mha(d_in, ii);
    blk[b].pair_tr = parse_tr(d_in, ii);
    blk[b].row = parse_mha(d_in, ii);
  }
  LinP conf1 = parse_lin(d_in, ii);
  LinP conf2 = parse_lin(d_in, ii);
  LinP coord1 = parse_lin(d_in, ii);
  LinP coord2 = parse_lin(d_in, ii);
  LinP dist1 = parse_lin(d_in, ii);
  LinP dist2 = parse_lin(d_in, ii);
  LinP msa_emb = parse_lin(d_in, ii);
  LinP pair2edge = parse_lin(d_in, ii);
  LinP pair_emb = parse_lin(d_in, ii);
  LinP single_proj = parse_lin(d_in, ii);
  StrP st[2];
  for (int l = 0; l < 2; ++l) {
    st[l].fu1 = parse_lin(d_in, ii);
    st[l].fu2 = parse_lin(d_in, ii);
    IpaP& I = st[l].ipa;
    I.eb = parse_lin(d_in, ii);
    I.gamma = (const float*)d_in[ii++];
    I.k = parse_lin(d_in, ii);
    I.kp = parse_lin(d_in, ii);
    I.out = parse_lin(d_in, ii);
    I.q = parse_lin(d_in, ii);
    I.qp = parse_lin(d_in, ii);
    I.v = parse_lin(d_in, ii);
    I.vp = parse_lin(d_in, ii);
  }

  // --------------------- workspace layout (16B-aligned) --------------------
  float* wp = (float*)d_ws;
  auto alloc = [&](size_t n) { n = (n + 3) & ~(size_t)3; float* r = wp; wp += n; return r; };
  float* msa   = alloc((size_t)MR * MSA_D);        // 768x256
  float* mc    = alloc((size_t)MR * MSA_D);        // col-transposed msa
  float* pair  = alloc((size_t)PR * PAIR_D);       // 2304x128
  float* qkv   = alloc((size_t)PR * 3 * PAIR_D > (size_t)MR * 3 * MSA_D
                           ? (size_t)PR * 3 * PAIR_D : (size_t)MR * 3 * MSA_D);
  float* attno = alloc((size_t)PR * PAIR_D);       // attention out (pre-proj)
  float* proj  = alloc((size_t)MR * MSA_D);        // col out-proj (s,n) order
  float* lnb   = alloc((size_t)PR * PAIR_D);       // layernorm / dist1 tmp
  float* hid   = alloc((size_t)PR * 4 * PAIR_D);   // FFN hidden / outer buf
  float* mp    = alloc((size_t)S * MSA_D);
  float* edge  = alloc((size_t)PR * 128);
  float* node  = alloc((size_t)S * NODE_D);
  float* qb    = alloc((size_t)S * IPA_H * NODE_D);
  float* kb    = alloc((size_t)S * IPA_H * NODE_D);
  float* vb    = alloc((size_t)S * IPA_H * NODE_D);
  float* qpb   = alloc((size_t)S * HP3);
  float* kpb   = alloc((size_t)S * HP3);
  float* vpb   = alloc((size_t)S * HP3);
  float* qg    = alloc((size_t)S * HP3);
  float* kg    = alloc((size_t)S * HP3);
  float* ebb   = alloc((size_t)PR * IPA_H);
  float* lg    = alloc((size_t)S * S * IPA_H);
  float* aw    = alloc((size_t)S * S * IPA_H);
  float* cat   = alloc((size_t)S * CATN);
  float* fub   = alloc((size_t)S * 12);
  float* frames= alloc((size_t)S * 12);
  float* t1    = alloc((size_t)S * NODE_D);

  auto ew = [&](int total) { return dim3(cdiv(total, 256)); };
  const float sc32  = 1.f / sqrtf(32.f);
  const float sc16  = 1.f / sqrtf(16.f);
  const float sc384 = 1.f / sqrtf((float)NODE_D);

  // ------------------------------ embeddings -------------------------------
  gemm(stream, msa_features, msa_emb.w, msa_emb.b, nullptr, msa, MR, MSA_D, VOCAB, 0);
  gemm(stream, pair_features, pair_emb.w, pair_emb.b, nullptr, pair, PR, PAIR_D, 256, 0);

  // ---------------------------- evoformer blocks ---------------------------
  for (int b = 0; b < 2; ++b) {
    const BlkP& P = blk[b];
    // row attention (16 seqs of length 48, E=256, H=8, dh=32)
    gemm(stream, msa, P.row.in_w, P.row.in_b, nullptr, qkv, MR, 3 * MSA_D, MSA_D, 0);
    mha_attn<32><<<ew(NSEQ * EVO_H * S), 256, 0, stream>>>(qkv, attno, NSEQ, S, EVO_H, sc32);
    gemm(stream, attno, P.row.out.w, P.row.out.b, msa, msa, MR, MSA_D, MSA_D, 0);
    // column attention (48 seqs of length 16)
    tr_ns<<<ew(MR * MSA_D), 256, 0, stream>>>(msa, mc, NSEQ, S, MSA_D);
    gemm(stream, mc, P.col.in_w, P.col.in_b, nullptr, qkv, MR, 3 * MSA_D, MSA_D, 0);
    mha_attn<32><<<ew(S * EVO_H * NSEQ), 256, 0, stream>>>(qkv, attno, S, NSEQ, EVO_H, sc32);
    gemm(stream, attno, P.col.out.w, P.col.out.b, nullptr, proj, MR, MSA_D, MSA_D, 0);
    add_tr<<<ew(MR * MSA_D), 256, 0, stream>>>(msa, proj, NSEQ, S, MSA_D);
    // MSA transformer
    layernorm_rows<<<ew(MR), 256, 0, stream>>>(msa, P.msa_tr.ln_g, P.msa_tr.ln_b, lnb, MR, MSA_D);
    gemm(stream, lnb, P.msa_tr.l1.w, P.msa_tr.l1.b, nullptr, hid, MR, 4 * MSA_D, MSA_D, 1);
    gemm(stream, hid, P.msa_tr.l2.w, P.msa_tr.l2.b, msa, msa, MR, MSA_D, 4 * MSA_D, 0);
    // pair attention (L=2304, H=8, dh=16)
    gemm(stream, pair, P.pair_attn.in_w, P.pair_attn.in_b, nullptr, qkv, PR, 3 * PAIR_D, PAIR_D, 0);
    mha_attn<16><<<ew(EVO_H * PR), 256, 0, stream>>>(qkv, attno, 1, PR, EVO_H, sc16);
    gemm(stream, attno, P.pair_attn.out.w, P.pair_attn.out.b, pair, pair, PR, PAIR_D, PAIR_D, 0);
    // pair transformer
    layernorm_rows<<<ew(PR), 256, 0, stream>>>(pair, P.pair_tr.ln_g, P.pair_tr.ln_b, lnb, PR, PAIR_D);
    gemm(stream, lnb, P.pair_tr.l1.w, P.pair_tr.l1.b, nullptr, hid, PR, 4 * PAIR_D, PAIR_D, 1);
    gemm(stream, hid, P.pair_tr.l2.w, P.pair_tr.l2.b, pair, pair, PR, PAIR_D, 4 * PAIR_D, 0);
    // outer product -> pair
    mean_seq<<<ew(S * MSA_D), 256, 0, stream>>>(msa, mp, NSEQ, S, MSA_D);
    outer_k<<<ew(PR * MSA_D), 256, 0, stream>>>(mp, hid, S, MSA_D);
    gemm(stream, hid, P.m2p.w, P.m2p.b, pair, pair, PR, PAIR_D, MSA_D, 0);
  }

  // --------------------------- structure module ----------------------------
  gemm(stream, msa, single_proj.w, single_proj.b, nullptr, node, S, NODE_D, MSA_D, 0); // msa[:,0]
  gemm(stream, pair, pair2edge.w, pair2edge.b, nullptr, edge, PR, 128, PAIR_D, 0);
  frame_init_k<<<ew(S * 12), 256, 0, stream>>>(frames, S);

  for (int l = 0; l < 2; ++l) {
    const IpaP& I = st[l].ipa;
    gemm(stream, node, I.q.w, I.q.b, nullptr, qb, S, IPA_H * NODE_D, NODE_D, 0);
    gemm(stream, node, I.k.w, I.k.b, nullptr, kb, S, IPA_H * NODE_D, NODE_D, 0);
    gemm(stream, node, I.v.w, I.v.b, nullptr, vb, S, IPA_H * NODE_D, NODE_D, 0);
    gemm(stream, node, I.qp.w, I.qp.b, nullptr, qpb, S, HP3, NODE_D, 0);
    gemm(stream, node, I.kp.w, I.kp.b, nullptr, kpb, S, HP3, NODE_D, 0);
    gemm(stream, node, I.vp.w, I.vp.b, nullptr, vpb, S, HP3, NODE_D, 0);
    frame_apply_k<<<ew(S * HP3), 256, 0, stream>>>(qpb, frames, qg, S, IPA_H * IPA_P);
    frame_apply_k<<<ew(S * HP3), 256, 0, stream>>>(kpb, frames, kg, S, IPA_H * IPA_P);
    gemm(stream, edge, I.eb.w, I.eb.b, nullptr, ebb, PR, IPA_H, 128, 0);
    ipa_logits_k<<<ew(S * S * IPA_H), 256, 0, stream>>>(qb, kb, qg, kg, ebb, I.gamma,
                                                        lg, S, IPA_H, NODE_D, HP3, sc384);
    softmax_over_j<<<ew(S * IPA_H), 256, 0, stream>>>(lg, aw, S, IPA_H);
    ipa_so_k<<<ew(S * IPA_H * NODE_D), 256, 0, stream>>>(aw, vb, cat, S, IPA_H, NODE_D, CATN);
    ipa_po_k<<<ew(S * HP3), 256, 0, stream>>>(aw, vpb, cat, S, IPA_H, HP3, CATN, IPA_H * NODE_D);
    gemm(stream, cat, I.out.w, I.out.b, nullptr, node, S, NODE_D, CATN, 0);
    // frame update
    gemm(stream, node, st[l].fu1.w, st[l].fu1.b, nullptr, t1, S, NODE_D, NODE_D, 1);
    gemm(stream, t1, st[l].fu2.w, st[l].fu2.b, nullptr, fub, S, 12, NODE_D, 0);
    frame_update_k<<<ew(S), 256, 0, stream>>>(frames, fub, S);
  }

  // -------------------------------- heads ----------------------------------
  float* out = (float*)d_out;
  float* out_coords = out;                       // 48*42  = 2016
  float* out_conf   = out_coords + 48 * 42;      // 48
  float* out_frames = out_conf + 48;             // 576
  float* out_dist   = out_frames + 576;          // 2304*64 = 147456
  float* out_pair   = out_dist + PR * 64;        // 294912
  float* out_node   = out_pair + PR * PAIR_D;    // 18432

  gemm(stream, node, coord1.w, coord1.b, nullptr, t1, S, NODE_D, NODE_D, 1);
  gemm(stream, t1, coord2.w, coord2.b, nullptr, out_coords, S, 42, NODE_D, 0);

  gemm(stream, node, conf1.w, conf1.b, nullptr, t1, S, NODE_D / 2, NODE_D, 1);
  gemm(stream, t1, conf2.w, conf2.b, nullptr, out_conf, S, 1, NODE_D / 2, 2);

  copy_k<<<ew(S * 12), 256, 0, stream>>>(frames, out_frames, S * 12);

  gemm(stream, pair, dist1.w, dist1.b, nullptr, lnb, PR, PAIR_D, PAIR_D, 1);
  gemm(stream, lnb, dist2.w, dist2.b, nullptr, out_dist, PR, 64, PAIR_D, 0);

  copy_k<<<ew(PR * PAIR_D), 256, 0, stream>>>(pair, out_pair, PR * PAIR_D);
  copy_k<<<ew(S * NODE_D), 256, 0, stream>>>(node, out_node, S * NODE_D);
}